// NetVLAD_81655918231660
// MI455X (gfx1250) — compile-verified
//
#include <hip/hip_runtime.h>
#include <hip/hip_bf16.h>
#include <math.h>

typedef __attribute__((ext_vector_type(2))) float v2f;
typedef __attribute__((ext_vector_type(8))) float v8f;

#define B_SZ 32
#define N_SZ 4096
#define C_SZ 256
#define K_SZ 64
#define D_SZ 256
#define KC_SZ (K_SZ * C_SZ)   // 16384
#define NTOK (B_SZ * N_SZ)    // 131072

__device__ __forceinline__ float waveReduceSum(float v) {
#pragma unroll
    for (int m = 16; m >= 1; m >>= 1) v += __shfl_xor(v, m, 32);
    return v;
}

__device__ __forceinline__ v8f wmma_f32(v2f a, v2f b, v8f c) {
    // V_WMMA_F32_16X16X4_F32 : D(16x16 f32) = A(16x4) * B(4x16) + C
    return __builtin_amdgcn_wmma_f32_16x16x4_f32(
        /*neg_a=*/false, a, /*neg_b=*/false, b,
        /*c_mod=*/(short)0, c, /*reuse_a=*/false, /*reuse_b=*/false);
}

// ---------------------------------------------------------------------------
// Kernel 1: per-token L2 normalization over C. One wave (32 lanes) per token,
// 8 floats per lane (2x float4). Streams x (134 MB) from HBM once; result
// stays resident in the 192 MB L2 for the two GEMMs that re-read it.
// ---------------------------------------------------------------------------
__global__ void k_normalize(const float* __restrict__ x, float* __restrict__ xn) {
    int tok  = blockIdx.x * (blockDim.x >> 5) + (threadIdx.x >> 5);
    int lane = threadIdx.x & 31;
    const float4* src = (const float4*)(x + (size_t)tok * C_SZ);
    float4* dst       = (float4*)(xn + (size_t)tok * C_SZ);
    float4 a = src[lane * 2 + 0];
    float4 b = src[lane * 2 + 1];
    float ss = a.x * a.x + a.y * a.y + a.z * a.z + a.w * a.w
             + b.x * b.x + b.y * b.y + b.z * b.z + b.w * b.w;
    ss = waveReduceSum(ss);
    float rn = 1.0f / fmaxf(sqrtf(ss), 1e-12f);
    a.x *= rn; a.y *= rn; a.z *= rn; a.w *= rn;
    b.x *= rn; b.y *= rn; b.z *= rn; b.w *= rn;
    dst[lane * 2 + 0] = a;
    dst[lane * 2 + 1] = b;
}

// ---------------------------------------------------------------------------
// Kernel 2: assignment GEMM logits[tok, k] = xn[tok,:] . conv_w[k,:]
// (WMMA f32 16x16x4, one wave per 16-token group, all 64 clusters = 4 tiles),
// fused softmax over the 64 clusters.
// ---------------------------------------------------------------------------
__global__ void k_assign(const float* __restrict__ xn,
                         const float* __restrict__ conv_w,
                         float* __restrict__ sa) {
    int g    = blockIdx.x * (blockDim.x >> 5) + (threadIdx.x >> 5); // 16-token group
    int lane = threadIdx.x & 31;
    int half = lane >> 4;
    int lm   = lane & 15;

    v8f zv = {0.f, 0.f, 0.f, 0.f, 0.f, 0.f, 0.f, 0.f};
    v8f acc[4] = {zv, zv, zv, zv};

    size_t arow = ((size_t)g * 16 + lm) * C_SZ + 2 * half;
    for (int c0 = 0; c0 < C_SZ; c0 += 4) {
        v2f a = *(const v2f*)(xn + arow + c0);
#pragma unroll
        for (int t = 0; t < 4; ++t) {
            v2f b = *(const v2f*)(conv_w + ((size_t)(t * 16 + lm)) * C_SZ + c0 + 2 * half);
            acc[t] = wmma_f32(a, b, acc[t]);
        }
    }

    // softmax over 64 clusters for each of the 16 tokens.
    // D layout: VGPR r, lanes 0-15 -> row r, lanes 16-31 -> row r+8; col = lane%16.
#pragma unroll
    for (int r = 0; r < 8; ++r) {
        float v0 = acc[0][r], v1 = acc[1][r], v2 = acc[2][r], v3 = acc[3][r];
        float mx = fmaxf(fmaxf(v0, v1), fmaxf(v2, v3));
#pragma unroll
        for (int m = 8; m >= 1; m >>= 1) mx = fmaxf(mx, __shfl_xor(mx, m, 32));
        float e0 = __expf(v0 - mx), e1 = __expf(v1 - mx);
        float e2 = __expf(v2 - mx), e3 = __expf(v3 - mx);
        float s = e0 + e1 + e2 + e3;
#pragma unroll
        for (int m = 8; m >= 1; m >>= 1) s += __shfl_xor(s, m, 32);
        float inv = 1.0f / s;
        size_t row = ((size_t)g * 16 + r + 8 * half) * K_SZ;
        sa[row + 0 * 16 + lm] = e0 * inv;
        sa[row + 1 * 16 + lm] = e1 * inv;
        sa[row + 2 * 16 + lm] = e2 * inv;
        sa[row + 3 * 16 + lm] = e3 * inv;
    }
}

// ---------------------------------------------------------------------------
// Kernel 3: asum[b,k] = sum_n sa[b,n,k], with speculative prefetch ahead.
// ---------------------------------------------------------------------------
__global__ void k_asum(const float* __restrict__ sa, float* __restrict__ asum) {
    __shared__ float lds[256];
    int b = blockIdx.x, t = threadIdx.x;
    int k = t & 63, p = t >> 6;
    float acc = 0.f;
    size_t base = (size_t)b * N_SZ + (size_t)p * 1024;
    for (int i0 = 0; i0 < 1024; i0 += 8) {
        __builtin_prefetch(sa + (base + i0 + 64) * K_SZ + k, 0, 1);
#pragma unroll
        for (int i = 0; i < 8; ++i)
            acc += sa[(base + i0 + i) * K_SZ + k];
    }
    lds[t] = acc;
    __syncthreads();
    if (t < 64) asum[b * K_SZ + t] = lds[t] + lds[t + 64] + lds[t + 128] + lds[t + 192];
}

// ---------------------------------------------------------------------------
// Kernel 4: VLAD GEMM vlad[b,k,c] += sum_n sa[b,n,k] * xn[b,n,c]
// Per batch: [64,4096]^T x [4096,256]. Split-K over tokens (4 chunks) with
// fp32 atomics. One wave per (b, m-tile, n-group-of-4, k-chunk); 8 waves/block.
// ---------------------------------------------------------------------------
__global__ void k_vlad(const float* __restrict__ xn, const float* __restrict__ sa,
                       float* __restrict__ vacc) {
    int w = blockIdx.x * (blockDim.x >> 5) + (threadIdx.x >> 5);
    int kc = w & 3, ng = (w >> 2) & 3, mt = (w >> 4) & 3, b = w >> 6;
    int lane = threadIdx.x & 31, half = lane >> 4, lm = lane & 15;

    v8f zv = {0.f, 0.f, 0.f, 0.f, 0.f, 0.f, 0.f, 0.f};
    v8f acc[4] = {zv, zv, zv, zv};
    size_t base = (size_t)b * N_SZ;

    for (int n0 = kc * 1024; n0 < kc * 1024 + 1024; n0 += 16) {
        // prefetch ~64 token-rows ahead (speculative; dropped if OOB)
        __builtin_prefetch(sa + (base + n0 + 64) * K_SZ + mt * 16 + lm, 0, 1);
        __builtin_prefetch(xn + (base + n0 + 64) * C_SZ + ng * 64 + lm, 0, 1);
#pragma unroll
        for (int ns = 0; ns < 16; ns += 4) {
            size_t nrow = base + n0 + ns + 2 * half;
            v2f a;
            a.x = sa[nrow * K_SZ + mt * 16 + lm];
            a.y = sa[(nrow + 1) * K_SZ + mt * 16 + lm];
#pragma unroll
            for (int tt = 0; tt < 4; ++tt) {
                int c0 = ng * 64 + tt * 16;
                v2f bb;
                bb.x = xn[nrow * C_SZ + c0 + lm];
                bb.y = xn[(nrow + 1) * C_SZ + c0 + lm];
                acc[tt] = wmma_f32(a, bb, acc[tt]);
            }
        }
    }
#pragma unroll
    for (int tt = 0; tt < 4; ++tt)
#pragma unroll
        for (int r = 0; r < 8; ++r) {
            int kk = mt * 16 + r + 8 * half;
            int cc = ng * 64 + tt * 16 + lm;
            atomicAdd(&vacc[((size_t)b * K_SZ + kk) * C_SZ + cc], acc[tt][r]);
        }
}

// ---------------------------------------------------------------------------
// Kernel 5: centroid subtract + intra-normalize each (b,k) row over C,
// accumulate global sum-of-squares per batch for the outer L2 norm.
// ---------------------------------------------------------------------------
__global__ void k_intranorm(const float* __restrict__ vacc, const float* __restrict__ asum,
                            const float* __restrict__ cent, float* __restrict__ vn,
                            float* __restrict__ gnorm) {
    int idx  = blockIdx.x * (blockDim.x >> 5) + (threadIdx.x >> 5); // (b,k) row
    int lane = threadIdx.x & 31;
    int b = idx >> 6, k = idx & 63;
    float sA = asum[idx];
    size_t rbase = (size_t)idx * C_SZ + lane * 8;
    size_t cbase = (size_t)k * C_SZ + lane * 8;
    float v[8];
    float ss = 0.f;
#pragma unroll
    for (int j = 0; j < 8; ++j) {
        float t = vacc[rbase + j] - sA * cent[cbase + j];
        v[j] = t;
        ss += t * t;
    }
    ss = waveReduceSum(ss);
    float rn = 1.0f / fmaxf(sqrtf(ss), 1e-12f);
#pragma unroll
    for (int j = 0; j < 8; ++j) vn[rbase + j] = v[j] * rn;
    if (lane == 0) atomicAdd(&gnorm[b], ss * rn * rn);
}

// ---------------------------------------------------------------------------
// Kernel 6: hidden GEMM [32 x 16384] x [16384 x 256] with WMMA, split-K=32,
// 8 waves/block, prefetching hidden_w (the 16.8 MB streamed operand).
// Global L2-norm scale (1/||vlad||) is applied later in the final kernel.
// ---------------------------------------------------------------------------
__global__ void k_hidden(const float* __restrict__ vn, const float* __restrict__ hw,
                         float* __restrict__ vh) {
    int w = blockIdx.x * (blockDim.x >> 5) + (threadIdx.x >> 5);
    int kc = w & 31, nt = (w >> 5) & 15, mt = w >> 9;
    int lane = threadIdx.x & 31, half = lane >> 4, lm = lane & 15;

    v8f acc = {0.f, 0.f, 0.f, 0.f, 0.f, 0.f, 0.f, 0.f};
    size_t arow = ((size_t)(mt * 16 + lm)) * KC_SZ + 2 * half;
    for (int k0 = kc * 512; k0 < kc * 512 + 512; k0 += 32) {
        // prefetch hidden_w ~64 k-rows ahead (speculative)
        __builtin_prefetch(hw + (size_t)(k0 + 64) * D_SZ + nt * 16 + lm, 0, 1);
#pragma unroll
        for (int ks = 0; ks < 32; ks += 4) {
            int k = k0 + ks;
            v2f a = *(const v2f*)(vn + arow + k);
            size_t kr = (size_t)(k + 2 * half);
            v2f bb;
            bb.x = hw[kr * D_SZ + nt * 16 + lm];
            bb.y = hw[(kr + 1) * D_SZ + nt * 16 + lm];
            acc = wmma_f32(a, bb, acc);
        }
    }
#pragma unroll
    for (int r = 0; r < 8; ++r) {
        int m = mt * 16 + r + 8 * half;
        atomicAdd(&vh[(size_t)m * D_SZ + nt * 16 + lm], acc[r]);
    }
}

// ---------------------------------------------------------------------------
// Kernel 7: global-norm scale -> BatchNorm -> gating GEMM -> BatchNorm ->
// sigmoid -> multiply. Single workgroup, 256 threads (one column each),
// 32 KB LDS for the normalized activations.
// ---------------------------------------------------------------------------
__global__ void k_final(const float* __restrict__ vh, const float* __restrict__ gnorm,
                        const float* __restrict__ gw,
                        const float* __restrict__ bn2g, const float* __restrict__ bn2b,
                        const float* __restrict__ gbng, const float* __restrict__ gbnb,
                        float* __restrict__ out) {
    __shared__ float lds_y[B_SZ * D_SZ];   // 32 KB
    __shared__ float s_invgn[B_SZ];
    int d = threadIdx.x;
    if (d < B_SZ) s_invgn[d] = 1.0f / fmaxf(sqrtf(gnorm[d]), 1e-12f);
    __syncthreads();

    // scale by 1/||vlad||_2 and BatchNorm over batch dim
    float sum = 0.f, sq = 0.f;
    for (int b = 0; b < B_SZ; ++b) {
        float yv = vh[(size_t)b * D_SZ + d] * s_invgn[b];
        lds_y[b * D_SZ + d] = yv;
        sum += yv; sq += yv * yv;
    }
    float m   = sum * (1.0f / B_SZ);
    float var = sq * (1.0f / B_SZ) - m * m;
    float inv = 1.0f / sqrtf(var + 1e-5f);
    float g2 = bn2g[d], b2 = bn2b[d];
    for (int b = 0; b < B_SZ; ++b)
        lds_y[b * D_SZ + d] = g2 * (lds_y[b * D_SZ + d] - m) * inv + b2;
    __syncthreads();

    // gating GEMM column d, then BN over batch + sigmoid + multiply
    float garr[B_SZ];
    float gs = 0.f, gq = 0.f;
    for (int b = 0; b < B_SZ; ++b) {
        float acc = 0.f;
        for (int dd = 0; dd < D_SZ; ++dd)
            acc += lds_y[b * D_SZ + dd] * gw[(size_t)dd * D_SZ + d];
        garr[b] = acc;
        gs += acc; gq += acc * acc;
    }
    float gm   = gs * (1.0f / B_SZ);
    float gvar = gq * (1.0f / B_SZ) - gm * gm;
    float ginv = 1.0f / sqrtf(gvar + 1e-5f);
    float gg = gbng[d], gb = gbnb[d];
    for (int b = 0; b < B_SZ; ++b) {
        float z = gg * (garr[b] - gm) * ginv + gb;
        float gate = 1.0f / (1.0f + __expf(-z));
        out[(size_t)b * D_SZ + d] = lds_y[b * D_SZ + d] * gate;
    }
}

// ---------------------------------------------------------------------------
extern "C" void kernel_launch(void* const* d_in, const int* in_sizes, int n_in,
                              void* d_out, int out_size, void* d_ws, size_t ws_size,
                              hipStream_t stream) {
    const float* x      = (const float*)d_in[0];
    const float* conv_w = (const float*)d_in[1];
    const float* cent   = (const float*)d_in[2];
    const float* hw     = (const float*)d_in[3];
    const float* gw     = (const float*)d_in[4];
    const float* bn2g   = (const float*)d_in[5];
    const float* bn2b   = (const float*)d_in[6];
    const float* gbng   = (const float*)d_in[7];
    const float* gbnb   = (const float*)d_in[8];
    float* out = (float*)d_out;

    char* ws = (char*)d_ws;
    const size_t XN_OFF   = 0;                                  // 134,217,728 B
    const size_t SA_OFF   = XN_OFF + (size_t)NTOK * C_SZ * 4;   // +33,554,432 B
    const size_t ASUM_OFF = SA_OFF + (size_t)NTOK * K_SZ * 4;   // +8,192 B
    const size_t VACC_OFF = ASUM_OFF + (size_t)B_SZ * K_SZ * 4; // +2,097,152 B
    const size_t GN_OFF   = VACC_OFF + (size_t)B_SZ * KC_SZ * 4;// +128 B
    const size_t VN_OFF   = GN_OFF + 128;                       // +2,097,152 B
    const size_t VH_OFF   = VN_OFF + (size_t)B_SZ * KC_SZ * 4;  // +32,768 B

    float* xn    = (float*)(ws + XN_OFF);
    float* sa    = (float*)(ws + SA_OFF);
    float* asum  = (float*)(ws + ASUM_OFF);
    float* vacc  = (float*)(ws + VACC_OFF);
    float* gnorm = (float*)(ws + GN_OFF);
    float* vn    = (float*)(ws + VN_OFF);
    float* vh    = (float*)(ws + VH_OFF);

    // zero accumulation buffers (graph-capture safe)
    hipMemsetAsync(vacc,  0, (size_t)B_SZ * KC_SZ * 4, stream);
    hipMemsetAsync(gnorm, 0, (size_t)B_SZ * 4, stream);
    hipMemsetAsync(vh,    0, (size_t)B_SZ * D_SZ * 4, stream);

    k_normalize<<<NTOK / 8, 256, 0, stream>>>(x, xn);
    k_assign<<<NTOK / 16 / 8, 256, 0, stream>>>(xn, conv_w, sa);
    k_asum<<<B_SZ, 256, 0, stream>>>(sa, asum);
    k_vlad<<<B_SZ * 4 * 4 * 4 / 8, 256, 0, stream>>>(xn, sa, vacc);
    k_intranorm<<<B_SZ * K_SZ / 8, 256, 0, stream>>>(vacc, asum, cent, vn, gnorm);
    k_hidden<<<2 * 16 * 32 / 8, 256, 0, stream>>>(vn, hw, vh);
    k_final<<<1, 256, 0, stream>>>(vh, gnorm, gw, bn2g, bn2b, gbng, gbnb, out);

    (void)in_sizes; (void)n_in; (void)out_size; (void)ws_size;
}